// Model_5351529250911
// MI455X (gfx1250) — compile-verified
//
#include <hip/hip_runtime.h>

#define T_STEPS 1000
#define IN_DIM  100
#define H_DIM   5000
#define O_DIM   100
#define ALPHA   0.1f
#define NOISE_IN 0.1f

typedef __attribute__((ext_vector_type(2))) float v2f;
typedef __attribute__((ext_vector_type(8))) float v8f;

__device__ __forceinline__ int imin(int a, int b) { return a < b ? a : b; }

// ---------------------------------------------------------------------------
// Kernel A: inj[t][h] = sum_k u[t][k] * W_in[h][k] + NOISE_IN * noise[t][h]
// fp32 WMMA 16x16x4.  8 waves/block share one M tile.  K-loop loads use
// clamped indices (branch-free); epilogue has a guard-free fast path for
// interior tiles so loads/stores can be clause'd.
// ---------------------------------------------------------------------------
__global__ void __launch_bounds__(256)
inj_gemm_wmma(const float* __restrict__ u,
              const float* __restrict__ noise,
              const float* __restrict__ W_in,
              float* __restrict__ inj) {
    const int wave = threadIdx.x >> 5;
    const int lane = threadIdx.x & 31;
    const int l15  = lane & 15;
    const int koff = (lane >> 4) * 2;            // 0 or 2

    const int n0 = (blockIdx.x * 8 + wave) * 16; // H tile
    const int m0 = blockIdx.y * 16;              // T tile

    const int mi = imin(m0 + l15, T_STEPS - 1);  // clamped A row
    const int ni = imin(n0 + l15, H_DIM - 1);    // clamped B col

    const float* __restrict__ ap = u    + (size_t)mi * IN_DIM + koff;
    const float* __restrict__ bp = W_in + (size_t)ni * IN_DIM + koff;

    v8f c = {};
    #pragma unroll 5
    for (int k = 0; k < IN_DIM; k += 4) {        // 25 WMMAs
        const v2f a = *(const v2f*)(ap + k);
        const v2f b = *(const v2f*)(bp + k);
        c = __builtin_amdgcn_wmma_f32_16x16x4_f32(false, a, false, b,
                                                  (short)0, c, false, false);
    }

    // C/D layout: lanes 0-15 -> rows m0..m0+7; lanes 16-31 -> m0+8..m0+15
    const int col   = n0 + l15;
    const int rbase = m0 + ((lane >> 4) << 3);

    if (m0 + 16 <= T_STEPS && n0 + 16 <= H_DIM) {
        // fast path: whole tile in range, no per-element guards
        const size_t base = (size_t)rbase * H_DIM + col;
        const float* __restrict__ np = noise + base;
        float*       __restrict__ ip = inj   + base;
        #pragma unroll
        for (int v = 0; v < 8; ++v)
            ip[(size_t)v * H_DIM] = c[v] + NOISE_IN * np[(size_t)v * H_DIM];
    } else if (col < H_DIM) {
        #pragma unroll
        for (int v = 0; v < 8; ++v) {
            const int row = rbase + v;
            if (row < T_STEPS) {
                inj[(size_t)row * H_DIM + col] =
                    c[v] + NOISE_IN * noise[(size_t)row * H_DIM + col];
            }
        }
    }
}

// ---------------------------------------------------------------------------
// Init: x = atanh(r0); hidden_states row 0 = r0
// ---------------------------------------------------------------------------
__global__ void init_state(const float* __restrict__ r0,
                           float* __restrict__ x,
                           float* __restrict__ hidden0) {
    const int h = blockIdx.x * blockDim.x + threadIdx.x;
    if (h < H_DIM) {
        const float rv = r0[h];
        x[h]       = atanhf(rv);
        hidden0[h] = rv;
    }
}

// ---------------------------------------------------------------------------
// One recurrent step.  r_t (20 KB) staged in LDS once per block, shared by all
// 8 waves; W_rec is the only global stream (100 MB/step, L2-resident since
// 100 MB < 192 MB L2).  One wave32 per row, float4 loads, 4 independent FMA
// chains, shuffle reduction, fused leaky-integrate + tanh.
// Grid divides exactly: 5000 rows = 625 blocks * 8 waves.
// ---------------------------------------------------------------------------
__global__ void __launch_bounds__(256)
rnn_step(const float* __restrict__ W_rec,
         const float* __restrict__ inj_t,
         const float* __restrict__ r_t,
         float* __restrict__ r_next,
         float* __restrict__ x) {
    __shared__ float4 rsh[H_DIM / 4];            // 20000 bytes

    const float4* __restrict__ rv4 = (const float4*)r_t;
    for (int i = threadIdx.x; i < H_DIM / 4; i += 256)
        rsh[i] = rv4[i];
    __syncthreads();

    const int wave = threadIdx.x >> 5;
    const int lane = threadIdx.x & 31;
    const int row  = blockIdx.x * 8 + wave;      // < 5000 always

    const float4* __restrict__ w4 = (const float4*)(W_rec + (size_t)row * H_DIM);

    float a0 = 0.f, a1 = 0.f, a2 = 0.f, a3 = 0.f;
    #pragma unroll 4
    for (int i = lane; i < H_DIM / 4; i += 32) { // 1250 float4 per row
        const float4 w = w4[i];
        const float4 r = rsh[i];
        a0 = fmaf(w.x, r.x, a0);
        a1 = fmaf(w.y, r.y, a1);
        a2 = fmaf(w.z, r.z, a2);
        a3 = fmaf(w.w, r.w, a3);
    }
    float acc = (a0 + a1) + (a2 + a3);

    #pragma unroll
    for (int off = 16; off > 0; off >>= 1)
        acc += __shfl_down(acc, off, 32);

    if (lane == 0) {
        const float xn = (1.0f - ALPHA) * x[row] + ALPHA * (acc + inj_t[row]);
        x[row]      = xn;
        r_next[row] = tanhf(xn);
    }
}

// ---------------------------------------------------------------------------
// Kernel C: o[t][j] = sum_h rs[t][h] * W_out[j][h]   (rs = hidden rows 1..T)
// 7 waves/block = all 7 O tiles of one T tile (shared rs rows -> L0 reuse).
// Clamped loads; fast-path epilogue for full tiles.
// ---------------------------------------------------------------------------
__global__ void __launch_bounds__(224)
out_gemm_wmma(const float* __restrict__ hidden,  // (T+1, H), rows 1..T = rs
              const float* __restrict__ W_out,   // (O, H) row-major
              float* __restrict__ o) {
    const int wave = threadIdx.x >> 5;           // 0..6 -> O tile
    const int lane = threadIdx.x & 31;
    const int l15  = lane & 15;
    const int koff = (lane >> 4) * 2;

    const int n0 = wave * 16;                    // O tile
    const int m0 = blockIdx.x * 16;              // T tile

    const int mi = imin(m0 + l15, T_STEPS - 1);
    const int ni = imin(n0 + l15, O_DIM - 1);

    const float* __restrict__ rs = hidden + H_DIM;  // skip row 0
    const float* __restrict__ ap = rs    + (size_t)mi * H_DIM + koff;
    const float* __restrict__ bp = W_out + (size_t)ni * H_DIM + koff;

    v8f c = {};
    #pragma unroll 5
    for (int k = 0; k < H_DIM; k += 4) {         // 1250 WMMAs
        const v2f a = *(const v2f*)(ap + k);
        const v2f b = *(const v2f*)(bp + k);
        c = __builtin_amdgcn_wmma_f32_16x16x4_f32(false, a, false, b,
                                                  (short)0, c, false, false);
    }

    const int col   = n0 + l15;
    const int rbase = m0 + ((lane >> 4) << 3);

    if (m0 + 16 <= T_STEPS && n0 + 16 <= O_DIM) {
        float* __restrict__ op = o + (size_t)rbase * O_DIM + col;
        #pragma unroll
        for (int v = 0; v < 8; ++v)
            op[(size_t)v * O_DIM] = c[v];
    } else if (col < O_DIM) {
        #pragma unroll
        for (int v = 0; v < 8; ++v) {
            const int row = rbase + v;
            if (row < T_STEPS) o[(size_t)row * O_DIM + col] = c[v];
        }
    }
}

// ---------------------------------------------------------------------------
extern "C" void kernel_launch(void* const* d_in, const int* in_sizes, int n_in,
                              void* d_out, int out_size, void* d_ws, size_t ws_size,
                              hipStream_t stream) {
    (void)in_sizes; (void)n_in; (void)out_size; (void)ws_size;

    const float* u     = (const float*)d_in[0];   // (T, I)
    const float* r0    = (const float*)d_in[1];   // (H,)
    const float* noise = (const float*)d_in[2];   // (T, H)
    const float* W_in  = (const float*)d_in[3];   // (H, I)
    const float* W_rec = (const float*)d_in[4];   // (H, H)
    const float* W_out = (const float*)d_in[5];   // (O, H)

    float* hidden = (float*)d_out;                             // (T+1, H)
    float* o      = hidden + (size_t)(T_STEPS + 1) * H_DIM;    // (T, O)

    float* inj = (float*)d_ws;                                 // (T, H) 20 MB
    float* x   = inj + (size_t)T_STEPS * H_DIM;                // (H,)

    // 1) inj = u @ W_in^T + 0.1*noise   (fp32 WMMA)
    {
        const int ntiles = (H_DIM + 15) / 16;                  // 313
        dim3 grid((ntiles + 7) / 8, (T_STEPS + 15) / 16);      // (40, 63)
        inj_gemm_wmma<<<grid, 256, 0, stream>>>(u, noise, W_in, inj);
    }

    // 2) x0 = atanh(r0); hidden[0] = r0
    init_state<<<(H_DIM + 255) / 256, 256, 0, stream>>>(r0, x, hidden);

    // 3) sequential scan: 1000 bandwidth-bound matvec steps (graph replays these)
    for (int t = 0; t < T_STEPS; ++t) {
        rnn_step<<<H_DIM / 8, 256, 0, stream>>>(
            W_rec,
            inj + (size_t)t * H_DIM,
            hidden + (size_t)t * H_DIM,        // r_t
            hidden + (size_t)(t + 1) * H_DIM,  // r_{t+1}
            x);
    }

    // 4) o = rs @ W_out^T   (fp32 WMMA, 7 O-tiles x 63 T-tiles)
    {
        dim3 grid((T_STEPS + 15) / 16);                        // 63 blocks, 7 waves
        out_gemm_wmma<<<grid, 224, 0, stream>>>(hidden, W_out, o);
    }
}